// GCN_12584254177620
// MI455X (gfx1250) — compile-verified
//
#include <hip/hip_runtime.h>
#include <stdint.h>

// ---------------------------------------------------------------------------
// 2-layer GCN forward for MI455X (gfx1250, wave32).
//
//   h1  = relu(dropout(x)) @ W1              (dropout fused into WMMA A-loads)
//   h2  = relu(dropout( dinv*(CSR-gather of dinv[s]*h1[s]) + h1*dinv^2 + b1 ))
//   h3  = h2 @ W2
//   out = dinv*(CSR-gather of dinv[s]*h3[s]) + h3*dinv^2 + b2
//
// Edge aggregation uses a device-built CSR (histogram -> scan -> permute) and
// pure gather loops: no atomics on the 205M-element hot path. GEMMs use
// V_WMMA_F32_16X16X4_F32 at full fp32 precision.
// ---------------------------------------------------------------------------

typedef __attribute__((ext_vector_type(2))) float v2f;
typedef __attribute__((ext_vector_type(8))) float v8f;

#define FEAT 128
#define HID  128
#define NCLS 40
#define SCAN_CHUNK 2048            // 256 threads * 8 elements

// ---------------- deterministic per-element hash for dropout ----------------
__device__ __forceinline__ float uhash01(uint32_t i, uint32_t seed) {
    uint32_t x = i ^ (seed * 0x9E3779B9u);
    x ^= x >> 16; x *= 0x7feb352du;
    x ^= x >> 15; x *= 0x846ca68bu;
    x ^= x >> 16;
    return (float)(x >> 8) * (1.0f / 16777216.0f);
}
__device__ __forceinline__ float drop_relu(float v, uint32_t idx, uint32_t seed) {
    float u = uhash01(idx, seed);
    float d = (u < 0.8f) ? v * 1.25f : 0.0f;
    return fmaxf(d, 0.0f);
}

// ======================= CSR construction ==================================
__global__ void __launch_bounds__(256)
k_zero_i32(int* __restrict__ p, int n) {
    int t = blockIdx.x * 256 + threadIdx.x;
    if (t < n) p[t] = 0;
}

__global__ void __launch_bounds__(256)
k_count(const int* __restrict__ dst, int* __restrict__ cnt, int E) {
    int t = blockIdx.x * 256 + threadIdx.x;
    if (t < E) atomicAdd(&cnt[dst[t]], 1);
}

__global__ void __launch_bounds__(256)
k_dinv(const int* __restrict__ cnt, float* __restrict__ dinv, int n) {
    int t = blockIdx.x * 256 + threadIdx.x;
    if (t < n) dinv[t] = __frsqrt_rn((float)cnt[t] + 1.0f);
}

// block-local exclusive scan of 2048 counts; emits per-block sum
__global__ void __launch_bounds__(256)
k_scan_block(const int* __restrict__ cnt, int* __restrict__ part,
             int* __restrict__ blockSums, int n) {
    __shared__ int lds[256];
    const int base = blockIdx.x * SCAN_CHUNK;
    int excl_local[8];
    int run = 0;
#pragma unroll
    for (int j = 0; j < 8; ++j) {
        int idx = base + threadIdx.x * 8 + j;
        excl_local[j] = run;
        run += (idx < n) ? cnt[idx] : 0;
    }
    lds[threadIdx.x] = run;
    __syncthreads();
    for (int off = 1; off < 256; off <<= 1) {
        int v = lds[threadIdx.x];
        int add = (threadIdx.x >= off) ? lds[threadIdx.x - off] : 0;
        __syncthreads();
        lds[threadIdx.x] = v + add;
        __syncthreads();
    }
    int excl = (threadIdx.x == 0) ? 0 : lds[threadIdx.x - 1];
#pragma unroll
    for (int j = 0; j < 8; ++j) {
        int idx = base + threadIdx.x * 8 + j;
        if (idx < n) part[idx] = excl + excl_local[j];
    }
    if (threadIdx.x == 255) blockSums[blockIdx.x] = lds[255];
}

// exclusive scan of up to 256 block sums (single block)
__global__ void __launch_bounds__(256)
k_scan_tops(int* __restrict__ blockSums, int nb) {
    __shared__ int lds[256];
    const int t = threadIdx.x;
    lds[t] = (t < nb) ? blockSums[t] : 0;
    __syncthreads();
    for (int off = 1; off < 256; off <<= 1) {
        int v = lds[t];
        int add = (t >= off) ? lds[t - off] : 0;
        __syncthreads();
        lds[t] = v + add;
        __syncthreads();
    }
    if (t < nb) blockSums[t] = (t == 0) ? 0 : lds[t - 1];
}

__global__ void __launch_bounds__(256)
k_scan_add(int* __restrict__ rowptr, const int* __restrict__ blockSums,
           int n, int E) {
    int t = blockIdx.x * 256 + threadIdx.x;
    if (t < n) rowptr[t] += blockSums[t / SCAN_CHUNK];
    if (t == 0) rowptr[n] = E;
}

__global__ void __launch_bounds__(256)
k_copy_i32(const int* __restrict__ a, int* __restrict__ b, int n) {
    int t = blockIdx.x * 256 + threadIdx.x;
    if (t < n) b[t] = a[t];
}

// bucket-fill: perm_src holds the src node of every in-edge, grouped by dst
__global__ void __launch_bounds__(256)
k_fill(const int* __restrict__ src, const int* __restrict__ dst,
       int* __restrict__ cursor, int* __restrict__ perm_src, int E) {
    int e = blockIdx.x * 256 + threadIdx.x;
    if (e >= E) return;
    int pos = atomicAdd(&cursor[dst[e]], 1);
    perm_src[pos] = src[e];
}

// ======================= GEMM 1 (fused dropout+relu on A) ===================
// C[M,128] = relu(dropout(X))[M,128] @ W[128,128].  One wave per 16-row strip
// spanning all 8 N-tiles; 32 K-steps of V_WMMA_F32_16X16X4_F32.
__global__ void __launch_bounds__(256)
k_gemm128_dr(const float* __restrict__ X, const float* __restrict__ W,
             float* __restrict__ C, int Mtiles, uint32_t seed) {
    const int wave  = threadIdx.x >> 5;
    const int lane  = threadIdx.x & 31;
    const int tileM = blockIdx.x * 8 + wave;
    if (tileM >= Mtiles) return;              // wave-uniform: EXEC all-1s

    const int lrow  = lane & 15;
    const int khalf = (lane >> 4) << 1;       // 0 or 2 (ISA A 16x4 f32 layout)
    const int row   = tileM * 16 + lrow;
    const float* arow = X + (size_t)row * FEAT;
    const uint32_t abase = (uint32_t)row * FEAT;

    v8f acc[8] = {};
    for (int k0 = 0; k0 < FEAT; k0 += 4) {
        v2f a;
        a.x = drop_relu(arow[k0 + khalf],     abase + k0 + khalf,     seed);
        a.y = drop_relu(arow[k0 + khalf + 1], abase + k0 + khalf + 1, seed);
        const float* w0 = W + (size_t)(k0 + khalf) * HID;
        const float* w1 = w0 + HID;
#pragma unroll
        for (int nt = 0; nt < 8; ++nt) {
            v2f b;
            b.x = w0[nt * 16 + lrow];
            b.y = w1[nt * 16 + lrow];
            acc[nt] = __builtin_amdgcn_wmma_f32_16x16x4_f32(
                false, a, false, b, (short)0, acc[nt], false, false);
        }
    }
    const int rowHi = (lane >> 4) << 3;       // 0 or 8 (C/D layout)
#pragma unroll
    for (int nt = 0; nt < 8; ++nt) {
        float* crow = C + (size_t)(tileM * 16 + rowHi) * HID + nt * 16 + lrow;
#pragma unroll
        for (int i = 0; i < 8; ++i)
            crow[(size_t)i * HID] = acc[nt][i];
    }
}

// ======================= GEMM 2: C[M,40] = A[M,128] @ W2[128,40] ============
__global__ void __launch_bounds__(256)
k_gemm40(const float* __restrict__ A, const float* __restrict__ W,
         float* __restrict__ C, int Mtiles) {
    const int wave  = threadIdx.x >> 5;
    const int lane  = threadIdx.x & 31;
    const int tileM = blockIdx.x * 8 + wave;
    if (tileM >= Mtiles) return;

    const int lrow  = lane & 15;
    const int khalf = (lane >> 4) << 1;
    const float* arow = A + (size_t)(tileM * 16 + lrow) * HID;

    v8f acc[3] = {};
    for (int k0 = 0; k0 < HID; k0 += 4) {
        v2f a;
        a.x = arow[k0 + khalf];
        a.y = arow[k0 + khalf + 1];
        const float* w0 = W + (size_t)(k0 + khalf) * NCLS;
        const float* w1 = w0 + NCLS;
#pragma unroll
        for (int nt = 0; nt < 3; ++nt) {
            int col = nt * 16 + lrow;
            v2f b;
            b.x = (col < NCLS) ? w0[col] : 0.0f;
            b.y = (col < NCLS) ? w1[col] : 0.0f;
            acc[nt] = __builtin_amdgcn_wmma_f32_16x16x4_f32(
                false, a, false, b, (short)0, acc[nt], false, false);
        }
    }
    const int rowHi = (lane >> 4) << 3;
#pragma unroll
    for (int nt = 0; nt < 3; ++nt) {
        int col = nt * 16 + lrow;
        if (col < NCLS) {
            float* crow = C + (size_t)(tileM * 16 + rowHi) * NCLS + col;
#pragma unroll
            for (int i = 0; i < 8; ++i)
                crow[(size_t)i * NCLS] = acc[nt][i];
        }
    }
}

// ========== gather layer 1 + epilogue + dropout2 + relu =====================
// h2[i,:] = relu(dropout( dinv[i]*sum_e dinv[s_e]*h1[s_e,:] + h1[i,:]*dinv^2
//                         + b1 ))
// One wave per node; lanes own float4 feature chunks; sources broadcast via
// __shfl; next rows software-prefetched (global_prefetch_b8).
__global__ void __launch_bounds__(256)
k_gather128(const float* __restrict__ h1, const float* __restrict__ dinv,
            const int* __restrict__ rowptr, const int* __restrict__ perm_src,
            const float* __restrict__ b1, float* __restrict__ h2,
            int N, uint32_t seed) {
    const int wave = threadIdx.x >> 5;
    const int lane = threadIdx.x & 31;
    const int i = blockIdx.x * 8 + wave;
    if (i >= N) return;                       // wave-uniform
    const int start = rowptr[i], end = rowptr[i + 1];

    float4 acc = {0.0f, 0.0f, 0.0f, 0.0f};
    for (int p0 = start; p0 < end; p0 += 32) {
        int myp = p0 + lane;
        int s_lane = (myp < end) ? perm_src[myp] : 0;
        float w_lane = (myp < end) ? dinv[s_lane] : 0.0f;
        __builtin_prefetch(h1 + (size_t)s_lane * HID, 0, 0);
        int m = min(32, end - p0);
        for (int j = 0; j < m; ++j) {
            int s   = __shfl(s_lane, j, 32);
            float w = __shfl(w_lane, j, 32);
            float4 v = ((const float4*)(h1 + (size_t)s * HID))[lane];
            acc.x += w * v.x; acc.y += w * v.y;
            acc.z += w * v.z; acc.w += w * v.w;
        }
    }
    const float di  = dinv[i];
    const float di2 = di * di;
    float4 self = ((const float4*)(h1 + (size_t)i * HID))[lane];
    float av[4] = {acc.x, acc.y, acc.z, acc.w};
    float sv[4] = {self.x, self.y, self.z, self.w};
    float ov[4];
#pragma unroll
    for (int j = 0; j < 4; ++j) {
        int c = lane * 4 + j;
        float v = av[j] * di + sv[j] * di2 + b1[c];
        ov[j] = drop_relu(v, (uint32_t)i * HID + c, seed);
    }
    float4 r = {ov[0], ov[1], ov[2], ov[3]};
    ((float4*)(h2 + (size_t)i * HID))[lane] = r;
}

// ========== gather layer 2 + epilogue ======================================
// out[i,c] = dinv[i]*sum_e dinv[s_e]*h3[s_e,c] + h3[i,c]*dinv^2 + b2[c]
// Lanes cover c = lane (all 32) and c = 32+lane (lanes 0..7).
__global__ void __launch_bounds__(256)
k_gather40(const float* __restrict__ h3, const float* __restrict__ dinv,
           const int* __restrict__ rowptr, const int* __restrict__ perm_src,
           const float* __restrict__ b2, float* __restrict__ out, int N) {
    const int wave = threadIdx.x >> 5;
    const int lane = threadIdx.x & 31;
    const int i = blockIdx.x * 8 + wave;
    if (i >= N) return;
    const int start = rowptr[i], end = rowptr[i + 1];
    const bool has2 = lane < 8;

    float a0 = 0.0f, a1 = 0.0f;
    for (int p0 = start; p0 < end; p0 += 32) {
        int myp = p0 + lane;
        int s_lane = (myp < end) ? perm_src[myp] : 0;
        float w_lane = (myp < end) ? dinv[s_lane] : 0.0f;
        __builtin_prefetch(h3 + (size_t)s_lane * NCLS, 0, 0);
        int m = min(32, end - p0);
        for (int j = 0; j < m; ++j) {
            int s   = __shfl(s_lane, j, 32);
            float w = __shfl(w_lane, j, 32);
            const float* hr = h3 + (size_t)s * NCLS;
            a0 += w * hr[lane];
            if (has2) a1 += w * hr[32 + lane];
        }
    }
    const float di  = dinv[i];
    const float di2 = di * di;
    const float* hs = h3 + (size_t)i * NCLS;
    float* orow = out + (size_t)i * NCLS;
    orow[lane] = a0 * di + hs[lane] * di2 + b2[lane];
    if (has2) {
        int c = 32 + lane;
        orow[c] = a1 * di + hs[c] * di2 + b2[c];
    }
}

// ---------------------------------------------------------------------------
extern "C" void kernel_launch(void* const* d_in, const int* in_sizes, int n_in,
                              void* d_out, int out_size, void* d_ws, size_t ws_size,
                              hipStream_t stream) {
    const float* x   = (const float*)d_in[0];
    const int*   ei  = (const int*)d_in[1];
    const float* W1  = (const float*)d_in[2];
    const float* b1  = (const float*)d_in[3];
    const float* W2  = (const float*)d_in[4];
    const float* b2  = (const float*)d_in[5];
    float*       out = (float*)d_out;

    const int N = in_sizes[0] / FEAT;          // 100000
    const int E = in_sizes[1] / 2;             // 1600000
    const int* src = ei;
    const int* dst = ei + E;

    // ---- workspace layout ----
    const size_t NH = (size_t)N * HID;         // 12.8M floats
    char* w = (char*)d_ws;
    float* h1       = (float*)w;                w += NH * sizeof(float);
    float* h2       = (float*)w;                w += NH * sizeof(float);
    float* h3       = (float*)w;                w += (size_t)N * NCLS * sizeof(float);
    float* dinv     = (float*)w;                w += (size_t)N * sizeof(float);
    int*   cnt      = (int*)w;                  w += (size_t)N * sizeof(int);
    int*   rowptr   = (int*)w;                  w += (size_t)(N + 1) * sizeof(int);
    int*   cursor   = (int*)w;                  w += (size_t)N * sizeof(int);
    int*   permsrc  = (int*)w;                  w += (size_t)E * sizeof(int);
    int*   blkSums  = (int*)w;

    const int Mtiles     = (N + 15) / 16;      // 6250 (exact for N=100000)
    const int gemmBlocks = (Mtiles + 7) / 8;
    const int nScanBlk   = (N + SCAN_CHUNK - 1) / SCAN_CHUNK;   // 49 <= 256
    const int gN   = (N + 255) / 256;
    const int gE   = (E + 255) / 256;
    const int gNod = (N + 7) / 8;              // one wave per node

    // ---- CSR build: histogram -> dinv -> scan -> bucket fill ----
    k_zero_i32 <<<gN, 256, 0, stream>>>(cnt, N);
    k_count    <<<gE, 256, 0, stream>>>(dst, cnt, E);
    k_dinv     <<<gN, 256, 0, stream>>>(cnt, dinv, N);
    k_scan_block<<<nScanBlk, 256, 0, stream>>>(cnt, rowptr, blkSums, N);
    k_scan_tops <<<1, 256, 0, stream>>>(blkSums, nScanBlk);
    k_scan_add  <<<gN, 256, 0, stream>>>(rowptr, blkSums, N, E);
    k_copy_i32  <<<gN, 256, 0, stream>>>(rowptr, cursor, N);
    k_fill      <<<gE, 256, 0, stream>>>(src, dst, cursor, permsrc, E);

    // ---- layer 1: fused dropout+relu GEMM, then fused gather/epilogue ----
    k_gemm128_dr<<<gemmBlocks, 256, 0, stream>>>(x, W1, h1, Mtiles, 0x1234567u);
    k_gather128 <<<gNod, 256, 0, stream>>>(h1, dinv, rowptr, permsrc, b1, h2,
                                           N, 0x89abcdefu);
    // ---- layer 2: GEMM then fused gather/epilogue into out ----
    k_gemm40    <<<gemmBlocks, 256, 0, stream>>>(h2, W2, h3, Mtiles);
    k_gather40  <<<gNod, 256, 0, stream>>>(h3, dinv, rowptr, permsrc, b2, out, N);

    (void)n_in; (void)out_size; (void)ws_size;
}